// ART_block_43095701848204
// MI455X (gfx1250) — compile-verified
//
#include <hip/hip_runtime.h>

// Problem constants (match reference: B, N, E, D = 8, 256, 65536, 512)
constexpr int Bn = 8;
constexpr int Nn = 256;
constexpr int En = 65536;
constexpr int Dn = 512;

typedef __attribute__((ext_vector_type(2))) float v2f;
typedef __attribute__((ext_vector_type(8))) float v8f;

// ---------------------------------------------------------------------------
// FP32 WMMA GEMM:  C[b] = act( A[b] (MxK)  x  B'  + bias (+ addmat) )
//   TRANSB=1 : B' = W^T where W is (Ncols x K) row-major   (b[k,n] = W[n,k])
//   TRANSB=0 : B' = Bm (K x Ncols) row-major               (b[k,n] = Bm[k,n])
// One wave32 per 16x16 output tile, V_WMMA_F32_16X16X4_F32 along K.
// A-frag layout : lanes 0-15 hold {K=k0,k0+1}, lanes 16-31 hold {K=k0+2,k0+3}
// C/D layout    : VGPR r, lanes 0-15 -> M=r, lanes 16-31 -> M=r+8 (N = lane&15)
// ---------------------------------------------------------------------------
template <bool TRANSB, bool RELU, bool ADDMAT>
__global__ void gemm_wmma_f32(const float* __restrict__ A,
                              const float* __restrict__ Bm,
                              const float* __restrict__ bias,
                              const float* __restrict__ addmat,
                              float* __restrict__ C,
                              int M, int K, int Ncols,
                              long strideA, long strideB, long strideC,
                              long strideAdd) {
  const int wave    = threadIdx.x >> 5;
  const int lane    = threadIdx.x & 31;
  const int tiles_n = Ncols >> 4;
  const int tile    = blockIdx.x * (blockDim.x >> 5) + wave;
  const int tm      = tile / tiles_n;
  const int tn      = tile - tm * tiles_n;
  const int b       = blockIdx.y;

  const float* Ab = A + (long)b * strideA;
  const float* Bb = Bm + (long)b * strideB;

  const int half = lane >> 4;   // 0: K=k0,k0+1   1: K=k0+2,k0+3
  const int l    = lane & 15;
  const int arow = (tm << 4) + l;   // A row owned by this lane
  const int bcol = (tn << 4) + l;   // B column owned by this lane

  v8f acc = {};
  for (int k0 = 0; k0 < K; k0 += 4) {
    const int ka = k0 + (half << 1);
    v2f a = *(const v2f*)(Ab + (long)arow * K + ka);
    v2f bf;
    if (TRANSB) {
      bf = *(const v2f*)(Bb + (long)bcol * K + ka);
    } else {
      bf.x = Bb[(long)ka * Ncols + bcol];
      bf.y = Bb[(long)(ka + 1) * Ncols + bcol];
    }
    acc = __builtin_amdgcn_wmma_f32_16x16x4_f32(false, a, false, bf,
                                                (short)0, acc, false, false);
  }

  const float bv = bias ? bias[bcol] : 0.0f;
#pragma unroll
  for (int r = 0; r < 8; ++r) {
    const int crow = (tm << 4) + r + (half << 3);
    float v = acc[r] + bv;
    if (ADDMAT) v += addmat[(long)b * strideAdd + (long)crow * Ncols + bcol];
    if (RELU) v = fmaxf(v, 0.0f);
    C[(long)b * strideC + (long)crow * Ncols + bcol] = v;
  }
}

// ---------------------------------------------------------------------------
// Edge kernel (bandwidth-critical: streams phr_feats ~1.07 GB exactly once).
// One wave per edge: atten = sum_d s[i,d]*o[j,d]*union[e,d]*w_w[d] + w_b
// then atomic scatter-add into att_raw[b, i, j].
// ---------------------------------------------------------------------------
__global__ void edge_kernel(const float* __restrict__ s,
                            const float* __restrict__ o,
                            const float* __restrict__ un,
                            const int* __restrict__ pairs,
                            const float* __restrict__ w_w,
                            const float* __restrict__ w_b,
                            float* __restrict__ att_raw) {
  const long gw   = (long)(blockIdx.x * blockDim.x + threadIdx.x) >> 5;
  const int  lane = threadIdx.x & 31;
  const int  b    = (int)(gw / En);
  const int  e    = (int)(gw - (long)b * En);

  const long pbase = ((long)b * En + e) * 2;
  const int  i = pairs[pbase + 0];
  const int  j = pairs[pbase + 1];

  const float* srow = s + ((long)b * Nn + i) * Dn;
  const float* orow = o + ((long)b * Nn + j) * Dn;
  const float* urow = un + ((long)b * En + e) * Dn;

  float partial = 0.0f;
#pragma unroll
  for (int t = 0; t < Dn / 128; ++t) {   // Dn/(32 lanes * 4) = 4 iterations
    const int d = t * 128 + lane * 4;
    float4 sv = *(const float4*)(srow + d);
    float4 ov = *(const float4*)(orow + d);
    float4 uv = *(const float4*)(urow + d);
    float4 wv = *(const float4*)(w_w + d);
    partial += sv.x * ov.x * uv.x * wv.x;
    partial += sv.y * ov.y * uv.y * wv.y;
    partial += sv.z * ov.z * uv.z * wv.z;
    partial += sv.w * ov.w * uv.w * wv.w;
  }
#pragma unroll
  for (int off = 16; off; off >>= 1) partial += __shfl_xor(partial, off, 32);

  if (lane == 0)
    atomicAdd(att_raw + ((long)b * Nn + i) * Nn + j, partial + w_b[0]);
}

// ---------------------------------------------------------------------------
// LayerNorm: one wave per row of length D=512 (16 elements / lane).
// ---------------------------------------------------------------------------
__global__ void ln_rows(const float* __restrict__ x,
                        const float* __restrict__ g,
                        const float* __restrict__ bt,
                        float* __restrict__ y) {
  const long gw   = (long)(blockIdx.x * blockDim.x + threadIdx.x) >> 5;
  const int  lane = threadIdx.x & 31;
  const float* xr = x + gw * Dn;

  float4 v[4];
  float sum = 0.0f, sq = 0.0f;
#pragma unroll
  for (int t = 0; t < 4; ++t) {
    v[t] = *(const float4*)(xr + t * 128 + lane * 4);
    sum += v[t].x + v[t].y + v[t].z + v[t].w;
    sq += v[t].x * v[t].x + v[t].y * v[t].y + v[t].z * v[t].z + v[t].w * v[t].w;
  }
#pragma unroll
  for (int off = 16; off; off >>= 1) {
    sum += __shfl_xor(sum, off, 32);
    sq += __shfl_xor(sq, off, 32);
  }
  const float mu  = sum * (1.0f / Dn);
  const float var = sq * (1.0f / Dn) - mu * mu;
  const float rs  = rsqrtf(var + 1e-5f);

#pragma unroll
  for (int t = 0; t < 4; ++t) {
    const int d = t * 128 + lane * 4;
    float4 gg = *(const float4*)(g + d);
    float4 bb = *(const float4*)(bt + d);
    float4 ov;
    ov.x = (v[t].x - mu) * rs * gg.x + bb.x;
    ov.y = (v[t].y - mu) * rs * gg.y + bb.y;
    ov.z = (v[t].z - mu) * rs * gg.z + bb.z;
    ov.w = (v[t].w - mu) * rs * gg.w + bb.w;
    *(float4*)(y + gw * Dn + d) = ov;
  }
}

// ---------------------------------------------------------------------------
// Softmax over rows of att (length N=256) with -10000 on the diagonal.
// One wave per row (8 elements / lane).
// ---------------------------------------------------------------------------
__global__ void softmax_att(const float* __restrict__ raw,
                            float* __restrict__ out) {
  const long gw   = (long)(blockIdx.x * blockDim.x + threadIdx.x) >> 5;
  const int  lane = threadIdx.x & 31;
  const int  r    = (int)(gw & (Nn - 1));   // row within the NxN matrix
  const float* row = raw + gw * Nn;

  float4 v[2];
  float mx = -3.4e38f;
#pragma unroll
  for (int t = 0; t < 2; ++t) {
    const int c = t * 128 + lane * 4;
    v[t] = *(const float4*)(row + c);
    if (c + 0 == r) v[t].x -= 10000.0f;
    if (c + 1 == r) v[t].y -= 10000.0f;
    if (c + 2 == r) v[t].z -= 10000.0f;
    if (c + 3 == r) v[t].w -= 10000.0f;
    mx = fmaxf(mx, fmaxf(fmaxf(v[t].x, v[t].y), fmaxf(v[t].z, v[t].w)));
  }
#pragma unroll
  for (int off = 16; off; off >>= 1) mx = fmaxf(mx, __shfl_xor(mx, off, 32));

  float s = 0.0f;
#pragma unroll
  for (int t = 0; t < 2; ++t) {
    v[t].x = __expf(v[t].x - mx);
    v[t].y = __expf(v[t].y - mx);
    v[t].z = __expf(v[t].z - mx);
    v[t].w = __expf(v[t].w - mx);
    s += v[t].x + v[t].y + v[t].z + v[t].w;
  }
#pragma unroll
  for (int off = 16; off; off >>= 1) s += __shfl_xor(s, off, 32);
  const float inv = 1.0f / s;

#pragma unroll
  for (int t = 0; t < 2; ++t) {
    const int c = t * 128 + lane * 4;
    float4 ov = {v[t].x * inv, v[t].y * inv, v[t].z * inv, v[t].w * inv};
    *(float4*)(out + gw * Nn + c) = ov;
  }
}

// out = a + b (float4 elementwise)
__global__ void add4_kernel(const float* __restrict__ a,
                            const float* __restrict__ b,
                            float* __restrict__ c) {
  const long idx = (long)blockIdx.x * blockDim.x + threadIdx.x;
  const float4 av = ((const float4*)a)[idx];
  const float4 bv = ((const float4*)b)[idx];
  float4 cv = {av.x + bv.x, av.y + bv.y, av.z + bv.z, av.w + bv.w};
  ((float4*)c)[idx] = cv;
}

// ---------------------------------------------------------------------------
extern "C" void kernel_launch(void* const* d_in, const int* in_sizes, int n_in,
                              void* d_out, int out_size, void* d_ws,
                              size_t ws_size, hipStream_t stream) {
  const float* obj    = (const float*)d_in[0];
  const float* phr    = (const float*)d_in[1];
  const int*   pairs  = (const int*)d_in[2];
  const float* ws_w   = (const float*)d_in[3];
  const float* ws_b   = (const float*)d_in[4];
  const float* wo_w   = (const float*)d_in[5];
  const float* wo_b   = (const float*)d_in[6];
  const float* w_w    = (const float*)d_in[7];
  const float* w_b    = (const float*)d_in[8];
  const float* conv_w = (const float*)d_in[9];
  const float* conv_b = (const float*)d_in[10];
  const float* ln1_g  = (const float*)d_in[11];
  const float* ln1_b  = (const float*)d_in[12];
  const float* ln2_g  = (const float*)d_in[13];
  const float* ln2_b  = (const float*)d_in[14];
  const float* t1_w   = (const float*)d_in[15];
  const float* t1_b   = (const float*)d_in[16];
  const float* t2_w   = (const float*)d_in[17];
  const float* t2_b   = (const float*)d_in[18];

  const long SND = (long)Nn * Dn;          // per-batch stride for (N,D) tensors
  const long SNN = (long)Nn * Nn;          // per-batch stride for (N,N)
  const long SN2D = (long)Nn * 2 * Dn;     // per-batch stride for (N,2D)

  // Output tuple layout: refined (B,N,D) | att (B,N,N) | ctx (B,N,D)
  float* out_refined = (float*)d_out;
  float* out_att     = out_refined + (long)Bn * SND;
  float* out_ctx     = out_att + (long)Bn * SNN;

  // Scratch layout (~30 MB of f32)
  float* ws       = (float*)d_ws;
  float* s_buf    = ws;                               // B*N*D
  float* o_buf    = s_buf + (long)Bn * SND;           // B*N*D
  float* ln_buf   = o_buf + (long)Bn * SND;           // B*N*D (ln1 then ln2)
  float* feat_buf = ln_buf + (long)Bn * SND;          // B*N*D
  float* out_buf  = feat_buf + (long)Bn * SND;        // B*N*D (obj + ctx)
  float* u_buf    = out_buf + (long)Bn * SND;         // B*N*2D
  float* att_raw  = u_buf + (long)Bn * SN2D;          // B*N*N

  hipMemsetAsync(att_raw, 0, (size_t)Bn * SNN * sizeof(float), stream);

  const dim3 blk(256);  // 8 wave32s per block; 1 wave = one 16x16 WMMA tile

  // s = obj @ ws_w^T + ws_b ; o = obj @ wo_w^T + wo_b   (M=256,K=512,N=512)
  {
    const int tiles = (Nn / 16) * (Dn / 16);  // 512
    gemm_wmma_f32<true, false, false><<<dim3(tiles / 8, Bn), blk, 0, stream>>>(
        obj, ws_w, ws_b, nullptr, s_buf, Nn, Dn, Dn, SND, 0, SND, 0);
    gemm_wmma_f32<true, false, false><<<dim3(tiles / 8, Bn), blk, 0, stream>>>(
        obj, wo_w, wo_b, nullptr, o_buf, Nn, Dn, Dn, SND, 0, SND, 0);
  }

  // ln_buf = LN(obj)
  ln_rows<<<dim3(Bn * Nn / 8), blk, 0, stream>>>(obj, ln1_g, ln1_b, ln_buf);

  // feat = relu(LN(obj) @ conv_w^T + conv_b)
  {
    const int tiles = (Nn / 16) * (Dn / 16);
    gemm_wmma_f32<true, true, false><<<dim3(tiles / 8, Bn), blk, 0, stream>>>(
        ln_buf, conv_w, conv_b, nullptr, feat_buf, Nn, Dn, Dn, SND, 0, SND, 0);
  }

  // Edge scatter: streams phr_feats (1.07 GB) once — the bandwidth roofline.
  edge_kernel<<<dim3((Bn * En) / 8), blk, 0, stream>>>(s_buf, o_buf, phr, pairs,
                                                       w_w, w_b, att_raw);

  // att = softmax(att_raw - 10000*I) -> second output slice
  softmax_att<<<dim3(Bn * Nn / 8), blk, 0, stream>>>(att_raw, out_att);

  // ctx = att @ feat  (M=256, K=256, N=512) -> third output slice
  {
    const int tiles = (Nn / 16) * (Dn / 16);
    gemm_wmma_f32<false, false, false><<<dim3(tiles / 8, Bn), blk, 0, stream>>>(
        out_att, feat_buf, nullptr, nullptr, out_ctx, Nn, Nn, Dn, SNN, SND, SND, 0);
  }

  // out = obj + ctx
  add4_kernel<<<dim3((Bn * SND / 4) / 256), blk, 0, stream>>>(obj, out_ctx,
                                                              out_buf);

  // h = LN(out)
  ln_rows<<<dim3(Bn * Nn / 8), blk, 0, stream>>>(out_buf, ln2_g, ln2_b, ln_buf);

  // u = relu(h @ t1_w^T + t1_b)   (M=256, K=512, N=1024)
  {
    const int tiles = (Nn / 16) * (2 * Dn / 16);  // 1024
    gemm_wmma_f32<true, true, false><<<dim3(tiles / 8, Bn), blk, 0, stream>>>(
        ln_buf, t1_w, t1_b, nullptr, u_buf, Nn, Dn, 2 * Dn, SND, 0, SN2D, 0);
  }

  // refined = relu(out + u @ t2_w^T + t2_b)  (M=256, K=1024, N=512) -> slice 1
  {
    const int tiles = (Nn / 16) * (Dn / 16);
    gemm_wmma_f32<true, true, true><<<dim3(tiles / 8, Bn), blk, 0, stream>>>(
        u_buf, t2_w, t2_b, out_buf, out_refined, Nn, 2 * Dn, Dn, SN2D, 0, SND,
        SND);
  }
}